// QNetwork_62577673503673
// MI455X (gfx1250) — compile-verified
//
#include <hip/hip_runtime.h>
#include <math.h>

typedef __attribute__((ext_vector_type(2))) float v2f;
typedef __attribute__((ext_vector_type(8))) float v8f;
typedef unsigned int u32x4 __attribute__((ext_vector_type(4)));
typedef int i32x8 __attribute__((ext_vector_type(8)));
typedef int i32x4 __attribute__((ext_vector_type(4)));

#define KCH 32

// ---------------------------------------------------------------------------
// Fast aligned f32 WMMA GEMM: C[M,N] = act(A[M,K] @ B[K,N] + bias)
// Requires M%32==0 and K%32==0. Block = 256 threads (8 wave32).
// Block tile: 32 rows x 128 cols; each wave owns a 16-col slice and two
// 16x16 M-tiles (two independent WMMA accumulation chains).
// A chunk [32][32] staged in LDS (row stride 36 dwords: 16B-aligned rows,
// conflict-free column reads). use_tdm=1 routes the stage through the
// Tensor Data Mover (tensor_load_to_lds + s_wait_tensorcnt).
// ---------------------------------------------------------------------------
__global__ __launch_bounds__(256) void gemm_wmma_fast(
    const float* __restrict__ A, const float* __restrict__ B,
    const float* __restrict__ bias, float* __restrict__ C,
    int M, int N, int K, int act, int use_tdm)
{
    __shared__ float As[32][36];
    const int lane  = threadIdx.x & 31;
    const int wave  = threadIdx.x >> 5;
    const int m0    = blockIdx.y << 5;
    const int n0    = (blockIdx.x << 7) + (wave << 4);
    const int mrow  = lane & 15;
    const int khalf = lane >> 4;              // 0 or 1
    const int ncol  = n0 + mrow;
    const int sr    = threadIdx.x >> 3;       // stage row 0..31
    const int sc    = (threadIdx.x & 7) << 2; // stage col 0,4,..,28
    v8f acc0 = {}, acc1 = {};
    for (int k0 = 0; k0 < K; k0 += KCH) {
        if (use_tdm) {
            if (wave == 0) {
                // --- Tensor DMA descriptor: 32x32 f32 tile of A -> LDS ---
                unsigned lds_base = (unsigned)(size_t)(void*)&As[0][0];
                unsigned long long ga =
                    (unsigned long long)(const void*)(A + (size_t)m0 * K + k0);
                u32x4 g0;
                g0.x = 1u;                                    // count=1 (valid)
                g0.y = lds_base;                              // lds_addr
                g0.z = (unsigned)(ga & 0xFFFFFFFFu);          // global_addr lo
                g0.w = (unsigned)((ga >> 32) & 0x01FFFFFFu) | (2u << 30); // hi | type=2
                i32x8 g1;
                // data_size=4B(2), pad_enable, pad_interval=32dw(4), pad_amount=4dw(3)
                g1[0] = (2 << 16) | (1 << 20) | (4 << 22) | (3 << 25);
                g1[1] = (int)(((unsigned)K & 0xFFFFu) << 16);            // dim0 lo16
                g1[2] = (int)((((unsigned)K >> 16) & 0xFFFFu) |
                              (((unsigned)M & 0xFFFFu) << 16));          // dim0 hi | dim1 lo
                g1[3] = (int)((((unsigned)M >> 16) & 0xFFFFu) | (32u << 16)); // dim1 hi | tile_dim0
                g1[4] = 32;                                   // tile_dim1=32, tile_dim2=0
                g1[5] = K;                                    // dim0_stride lo32
                g1[6] = 0;
                g1[7] = 0;
                i32x4 z4 = {0, 0, 0, 0};
                i32x8 z8 = {0, 0, 0, 0, 0, 0, 0, 0};
                __builtin_amdgcn_tensor_load_to_lds(g0, g1, z4, z4, z8, 0);
                __builtin_amdgcn_s_wait_tensorcnt(0);
            }
        } else {
            const float4 av = *reinterpret_cast<const float4*>(
                A + (size_t)(m0 + sr) * K + (k0 + sc));
            *reinterpret_cast<float4*>(&As[sr][sc]) = av;
            if (k0 + KCH < K)
                __builtin_prefetch(A + (size_t)(m0 + sr) * K + (k0 + KCH + sc), 0, 1);
        }
        __syncthreads();
        if (n0 < N) {   // wave-uniform
            #pragma unroll
            for (int kk = 0; kk < KCH; kk += 4) {
                int ka = kk + 2 * khalf;
                v2f a0, a1, b;
                a0.x = As[mrow][ka];
                a0.y = As[mrow][ka + 1];
                a1.x = As[16 + mrow][ka];
                a1.y = As[16 + mrow][ka + 1];
                int kb = k0 + ka;
                b.x = B[(size_t)kb * N + ncol];
                b.y = B[(size_t)(kb + 1) * N + ncol];
                acc0 = __builtin_amdgcn_wmma_f32_16x16x4_f32(
                           false, a0, false, b, (short)0, acc0, false, false);
                acc1 = __builtin_amdgcn_wmma_f32_16x16x4_f32(
                           false, a1, false, b, (short)0, acc1, false, false);
            }
        }
        __syncthreads();
    }
    if (n0 < N) {
        float bv = bias ? bias[ncol] : 0.0f;
        #pragma unroll
        for (int i = 0; i < 8; ++i) {
            int r0 = m0 + i + (khalf << 3);
            int r1 = r0 + 16;
            float v0 = acc0[i] + bv;
            float v1 = acc1[i] + bv;
            if (act == 1)      { v0 = fmaxf(v0, 0.0f); v1 = fmaxf(v1, 0.0f); }
            else if (act == 2) { v0 = tanhf(v0);       v1 = tanhf(v1);       }
            C[(size_t)r0 * N + ncol] = v0;
            C[(size_t)r1 * N + ncol] = v1;
        }
    }
}

// ---------------------------------------------------------------------------
// Generic guarded GEMM for ragged shapes (K=28/129, M=8193). Guards are
// selects over clamped unconditional loads (no exec-mask branching).
// ---------------------------------------------------------------------------
__global__ __launch_bounds__(256) void gemm_wmma(
    const float* __restrict__ A, const float* __restrict__ B,
    const float* __restrict__ bias, float* __restrict__ C,
    int M, int N, int K, int act)
{
    __shared__ float As[16][KCH + 1];
    const int lane  = threadIdx.x & 31;
    const int wave  = threadIdx.x >> 5;
    const int m0    = blockIdx.y << 4;
    const int n0    = (blockIdx.x << 7) + (wave << 4);
    const int mrow  = lane & 15;
    const int khalf = lane >> 4;
    const int ncol  = n0 + mrow;
    const int ncolc = (ncol < N) ? ncol : (N - 1);
    v8f acc = {};
    for (int k0 = 0; k0 < K; k0 += KCH) {
        for (int idx = threadIdx.x; idx < 16 * KCH; idx += 256) {
            int r = idx >> 5, c = idx & 31;
            int gm = m0 + r, gk = k0 + c;
            int gmc = (gm < M) ? gm : (M - 1);
            int gkc = (gk < K) ? gk : (K - 1);
            float v = A[(size_t)gmc * K + gkc];
            As[r][c] = (gm < M && gk < K) ? v : 0.0f;
        }
        __syncthreads();
        if (n0 < N) {
            #pragma unroll
            for (int kk = 0; kk < KCH; kk += 4) {
                int ka = kk + 2 * khalf;
                v2f a, b;
                a.x = As[mrow][ka];
                a.y = As[mrow][ka + 1];
                int kb = k0 + ka;
                int kb0 = (kb     < K) ? kb     : (K - 1);
                int kb1 = (kb + 1 < K) ? kb + 1 : (K - 1);
                float bx = B[(size_t)kb0 * N + ncolc];
                float by = B[(size_t)kb1 * N + ncolc];
                b.x = (kb     < K) ? bx : 0.0f;
                b.y = (kb + 1 < K) ? by : 0.0f;
                acc = __builtin_amdgcn_wmma_f32_16x16x4_f32(
                          false, a, false, b, (short)0, acc, false, false);
            }
        }
        __syncthreads();
    }
    if (n0 < N) {
        float bv = bias ? bias[ncol] : 0.0f;
        #pragma unroll
        for (int i = 0; i < 8; ++i) {
            int row = m0 + i + (khalf << 3);
            if (row < M) {
                float v = acc[i] + bv;
                if (act == 1)      v = fmaxf(v, 0.0f);
                else if (act == 2) v = tanhf(v);
                C[(size_t)row * N + ncol] = v;
            }
        }
    }
}

// ---------------------------------------------------------------------------
// Persistent single-block RNN recurrence: h_t = tanh(XW[t] + h_{t-1} @ Whh)
// ---------------------------------------------------------------------------
__global__ __launch_bounds__(256) void rnn_recurrent(
    const float* __restrict__ XW, const float* __restrict__ Whh,
    const float* __restrict__ h0, float* __restrict__ Y,
    float* __restrict__ hT)
{
    __shared__ float hs[32][132];
    const int lane  = threadIdx.x & 31;
    const int wave  = threadIdx.x >> 5;
    const int mrow  = lane & 15;
    const int khalf = lane >> 4;
    const int ncol  = (wave << 4) + mrow;
    for (int i = threadIdx.x; i < 32 * 128; i += 256)
        hs[i >> 7][i & 127] = h0[i];
    __syncthreads();
    for (int t = 0; t < 64; ++t) {
        float nv[2][8];
        #pragma unroll
        for (int mt = 0; mt < 2; ++mt) {
            v8f acc = {};
            #pragma unroll
            for (int kk = 0; kk < 128; kk += 4) {
                int ka = kk + 2 * khalf;
                v2f a, b;
                a.x = hs[(mt << 4) + mrow][ka];
                a.y = hs[(mt << 4) + mrow][ka + 1];
                b.x = Whh[(size_t)ka * 128 + ncol];
                b.y = Whh[(size_t)(ka + 1) * 128 + ncol];
                acc = __builtin_amdgcn_wmma_f32_16x16x4_f32(
                          false, a, false, b, (short)0, acc, false, false);
            }
            #pragma unroll
            for (int i = 0; i < 8; ++i) {
                int row = (mt << 4) + i + (khalf << 3);
                nv[mt][i] = tanhf(acc[i] + XW[((size_t)row * 64 + t) * 128 + ncol]);
            }
        }
        __syncthreads();
        #pragma unroll
        for (int mt = 0; mt < 2; ++mt)
            #pragma unroll
            for (int i = 0; i < 8; ++i) {
                int row = (mt << 4) + i + (khalf << 3);
                hs[row][ncol] = nv[mt][i];
                Y[((size_t)row * 64 + t) * 128 + ncol] = nv[mt][i];
            }
        __syncthreads();
    }
    for (int i = threadIdx.x; i < 32 * 128; i += 256)
        hT[i] = hs[i >> 7][i & 127];
}

// ---------------------------------------------------------------------------
// Elementwise / reduction / scatter helpers
// ---------------------------------------------------------------------------
__global__ void k_fill0(float* p, int n) {
    int i = blockIdx.x * 256 + threadIdx.x;
    if (i < n) p[i] = 0.0f;
}
__global__ void k_add(float* a, const float* __restrict__ b, int n) {
    int i = blockIdx.x * 256 + threadIdx.x;
    if (i < n) a[i] += b[i];
}
__global__ void k_gin_comb(float* aggr, const float* __restrict__ x,
                           const float* __restrict__ epsp, int layer, int n) {
    int i = blockIdx.x * 256 + threadIdx.x;
    if (i < n) aggr[i] = (1.0f + epsp[layer]) * x[i] + aggr[i];
}
__global__ void k_max3(const float* __restrict__ a, const float* __restrict__ b,
                       const float* __restrict__ c, float* o, int n) {
    int i = blockIdx.x * 256 + threadIdx.x;
    if (i < n) o[i] = fmaxf(fmaxf(a[i], b[i]), c[i]);
}
__global__ void k_scat(const float* __restrict__ x, const int* __restrict__ src,
                       const int* __restrict__ dst, const float* __restrict__ w,
                       float* out, int E, int Dd) {
    int e = blockIdx.x * 8 + (threadIdx.x >> 5);
    int lane = threadIdx.x & 31;
    if (e >= E) return;
    float ww = w ? w[e] : 1.0f;
    const float* xs = x + (size_t)src[e] * Dd;
    float* od = out + (size_t)dst[e] * Dd;
    for (int c = lane; c < Dd; c += 32) atomicAdd(&od[c], ww * xs[c]);
}
__global__ void k_deg(const int* __restrict__ dst, const float* __restrict__ w,
                      float* deg, int E) {
    int i = blockIdx.x * 256 + threadIdx.x;
    if (i < E) atomicAdd(&deg[dst[i]], w[i]);
}
__global__ void k_dinv(const float* __restrict__ deg, float* dinv, int n) {
    int i = blockIdx.x * 256 + threadIdx.x;
    if (i < n) { float d = deg[i]; dinv[i] = (d > 0.0f) ? rsqrtf(d) : 0.0f; }
}
__global__ void k_norm(const int* __restrict__ src, const int* __restrict__ dst,
                       const float* __restrict__ w, const float* __restrict__ dinv,
                       float* nrm, int E) {
    int i = blockIdx.x * 256 + threadIdx.x;
    if (i < E) nrm[i] = dinv[src[i]] * w[i] * dinv[dst[i]];
}
__global__ __launch_bounds__(256) void k_bn_reduce(const float* __restrict__ X,
                                                   float* mean, float* var,
                                                   int M, int N) {
    int col = blockIdx.x;
    float s = 0.0f, ss = 0.0f;
    for (int r = threadIdx.x; r < M; r += 256) {
        float v = X[(size_t)r * N + col];
        s += v; ss += v * v;
    }
    __shared__ float sh[256], sh2[256];
    sh[threadIdx.x] = s; sh2[threadIdx.x] = ss;
    __syncthreads();
    for (int st = 128; st > 0; st >>= 1) {
        if (threadIdx.x < st) {
            sh[threadIdx.x]  += sh[threadIdx.x + st];
            sh2[threadIdx.x] += sh2[threadIdx.x + st];
        }
        __syncthreads();
    }
    if (threadIdx.x == 0) {
        float mu = sh[0] / (float)M;
        mean[col] = mu;
        var[col]  = sh2[0] / (float)M - mu * mu;
    }
}
__global__ void k_bn_apply(float* X, const float* __restrict__ mean,
                           const float* __restrict__ var,
                           const float* __restrict__ scale,
                           const float* __restrict__ bias,
                           int M, int N, int relu) {
    int i = blockIdx.x * 256 + threadIdx.x;
    if (i >= M * N) return;
    int col = i % N;
    float v = (X[i] - mean[col]) * rsqrtf(var[col] + 1e-5f) * scale[col] + bias[col];
    if (relu) v = fmaxf(v, 0.0f);
    X[i] = v;
}
__global__ void k_gather_rows(const float* __restrict__ tab, const int* __restrict__ idx,
                              float* out, int rows, int Dd) {
    int i = blockIdx.x * 256 + threadIdx.x;
    if (i >= rows * Dd) return;
    int r = i / Dd, c = i - r * Dd;
    out[i] = tab[(size_t)idx[r] * Dd + c];
}
__global__ void k_full_cat(const float* __restrict__ ein, const float* __restrict__ eout,
                           float* out, int n) {
    int i = blockIdx.x * 256 + threadIdx.x;
    if (i >= n) return;
    int r = i >> 8, c = i & 255;
    out[i] = (r == 0) ? 0.0f
             : ((c < 128) ? ein[(size_t)(r - 1) * 128 + c]
                          : eout[(size_t)(r - 1) * 128 + (c - 128)]);
}
__global__ void k_traces_emb(const float* __restrict__ traces,
                             const float* __restrict__ fge, float* out, int n) {
    int i = blockIdx.x * 256 + threadIdx.x;
    if (i >= n) return;
    int r = i / 129, c = i - r * 129;
    if (c < 128) {
        int tid = (int)traces[(size_t)r * 2];
        out[i] = fge[(size_t)tid * 128 + c];
    } else {
        out[i] = traces[(size_t)r * 2 + 1];
    }
}
__global__ void k_bias_sums(const float* __restrict__ ti, const float* __restrict__ th,
                            const float* __restrict__ si, const float* __restrict__ sh,
                            float* out) {
    int i = blockIdx.x * 256 + threadIdx.x;
    if (i < 384)       out[i] = ti[i] + th[i];
    else if (i < 768)  out[i] = si[i - 384] + sh[i - 384];
}
__global__ void k_comb(const float* __restrict__ te, const float* __restrict__ se,
                       float* out, int n) {
    int i = blockIdx.x * 256 + threadIdx.x;
    if (i >= n) return;
    int r = i >> 7, c = i & 127;
    out[i] = (c < 64) ? te[(size_t)r * 64 + c] : se[(size_t)r * 64 + (c - 64)];
}
__global__ void k_attn(const float* __restrict__ xp, const float* __restrict__ cp,
                       float* out, int n) {
    int i = blockIdx.x * 256 + threadIdx.x;
    if (i >= n) return;
    int bl = i / 10;
    const float* a = xp + (size_t)bl * 128;
    const float* b = cp + (size_t)i * 128;
    float s = 0.0f;
    for (int k = 0; k < 128; ++k) s += a[k] * b[k];
    out[i] = s;
}

// ---------------------------------------------------------------------------
static inline int cdiv(int a, int b) { return (a + b - 1) / b; }

extern "C" void kernel_launch(void* const* d_in, const int* in_sizes, int n_in,
                              void* d_out, int out_size, void* d_ws, size_t ws_size,
                              hipStream_t stream) {
    (void)in_sizes; (void)n_in; (void)out_size; (void)ws_size;
    const int NR = 16384, NG = 8192, ER = 262144, ET = 131072;
    const int D = 128, BL = 32 * 64;               // 2048

    const float* traces = (const float*)d_in[0];
    const int*   matched= (const int*)  d_in[1];
    const int*   cand   = (const int*)  d_in[2];
    const float* t_enc  = (const float*)d_in[3];
    const float* s_enc  = (const float*)d_in[4];
    const float* road_x = (const float*)d_in[5];
    const int*   redge  = (const int*)  d_in[6];
    const float* mapm   = (const float*)d_in[7];
    const int*   in_e   = (const int*)  d_in[8];
    const int*   out_e  = (const int*)  d_in[9];
    const float* ew     = (const float*)d_in[10];
    const float* Wrf    = (const float*)d_in[11];
    const float* brf    = (const float*)d_in[12];
    const float* geps   = (const float*)d_in[13];
    const float* gW1    = (const float*)d_in[14];
    const float* gb1    = (const float*)d_in[15];
    const float* gbn1s  = (const float*)d_in[16];
    const float* gbn1b  = (const float*)d_in[17];
    const float* gW2    = (const float*)d_in[18];
    const float* gb2    = (const float*)d_in[19];
    const float* gbns   = (const float*)d_in[20];
    const float* gbnb   = (const float*)d_in[21];
    const float* fcW    = (const float*)d_in[22];
    const float* fcb    = (const float*)d_in[23];
    const float* glinW  = (const float*)d_in[24];
    const float* gconvW = (const float*)d_in[25];
    const float* gcbns  = (const float*)d_in[26];
    const float* gcbnb  = (const float*)d_in[27];
    const float* tWih0  = (const float*)d_in[28];
    const float* tWih   = (const float*)d_in[29];
    const float* tWhh   = (const float*)d_in[30];
    const float* tbih   = (const float*)d_in[31];
    const float* tbhh   = (const float*)d_in[32];
    const float* sWih   = (const float*)d_in[33];
    const float* sWhh   = (const float*)d_in[34];
    const float* sbih   = (const float*)d_in[35];
    const float* sbhh   = (const float*)d_in[36];
    const float* twW    = (const float*)d_in[37];
    const float* twb    = (const float*)d_in[38];
    const float* swW    = (const float*)d_in[39];
    const float* swb    = (const float*)d_in[40];
    const float* attW   = (const float*)d_in[41];
    const float* attb   = (const float*)d_in[42];
    const float* atcW   = (const float*)d_in[43];
    const float* atcb   = (const float*)d_in[44];

    float* outp = (float*)d_out;
    float* th_o = outp;                 // traces_hidden  [3,32,128]
    float* sh_o = outp + 12288;         // segments_hidden[3,32,128]
    float* av_o = outp + 24576;         // action_values  [32,64,10]
    float* re_o = outp + 45056;         // road_emb       [16385,128]
    float* fg_o = outp + 2142336;       // full_grid_emb  [8193,128]

    float* ws = (float*)d_ws;
    size_t off = 0;
    auto alloc = [&](size_t n) { float* p = ws + off; off += (n + 63) & ~(size_t)63; return p; };
    auto gemm_g = [&](const float* A, const float* Bm, const float* bias, float* Cm,
                      int M, int N, int K, int act) {
        dim3 g(cdiv(N, 128), cdiv(M, 16));
        gemm_wmma<<<g, 256, 0, stream>>>(A, Bm, bias, Cm, M, N, K, act);
    };
    auto gemm_f = [&](const float* A, const float* Bm, const float* bias, float* Cm,
                      int M, int N, int K, int act, int tdm) {
        dim3 g(cdiv(N, 128), M / 32);
        gemm_wmma_fast<<<g, 256, 0, stream>>>(A, Bm, bias, Cm, M, N, K, act, tdm);
    };
    auto fill0 = [&](float* p, int n) { k_fill0<<<cdiv(n, 256), 256, 0, stream>>>(p, n); };

    // ================= Phase 1: road GIN =================
    float* xbuf = alloc((size_t)NR * D);
    float* aggr = alloc((size_t)NR * D);
    float* h1   = alloc((size_t)NR * 2 * D);
    float* outs = alloc((size_t)3 * NR * D);
    float* mn   = alloc(256);
    float* vr   = alloc(256);

    gemm_g(road_x, Wrf, brf, xbuf, NR, D, 28, 0);             // K=28 -> generic
    const int* rsrc = redge;
    const int* rdst = redge + ER;
    const float* xcur = xbuf;
    for (int l = 0; l < 3; ++l) {
        float* ol = outs + (size_t)l * NR * D;
        fill0(aggr, NR * D);
        k_scat<<<cdiv(ER, 8), 256, 0, stream>>>(xcur, rsrc, rdst, nullptr, aggr, ER, D);
        k_gin_comb<<<cdiv(NR * D, 256), 256, 0, stream>>>(aggr, xcur, geps, l, NR * D);
        gemm_f(aggr, gW1 + (size_t)l * D * 2 * D, gb1 + l * 256, h1, NR, 2 * D, D, 0, 0);
        k_bn_reduce<<<256, 256, 0, stream>>>(h1, mn, vr, NR, 256);
        k_bn_apply<<<cdiv(NR * 256, 256), 256, 0, stream>>>(h1, mn, vr,
                     gbn1s + l * 256, gbn1b + l * 256, NR, 256, 1);
        gemm_f(h1, gW2 + (size_t)l * 2 * D * D, gb2 + l * D, ol, NR, D, 2 * D, 0, 0);
        k_bn_reduce<<<128, 256, 0, stream>>>(ol, mn, vr, NR, D);
        k_bn_apply<<<cdiv(NR * D, 256), 256, 0, stream>>>(ol, mn, vr,
                     gbns + l * D, gbnb + l * D, NR, D, 1);
        xcur = ol;
    }
    k_max3<<<cdiv(NR * D, 256), 256, 0, stream>>>(outs, outs + (size_t)NR * D,
                     outs + (size_t)2 * NR * D, re_o, NR * D);
    fill0(re_o + (size_t)NR * D, D);                          // appended zero row

    // ================= Phase 2: grids, RNNs, attention (reuse ws) =================
    off = 0;
    float* pure_grid = alloc((size_t)NG * D);
    float* deg   = alloc(NG);
    float* dinv  = alloc(NG);
    float* nrm   = alloc(ET);
    float* hrbuf = alloc((size_t)NG * D);
    float* xlbuf = alloc((size_t)NG * D);
    float* ga[2] = { alloc((size_t)NG * D), alloc((size_t)NG * D) };
    float* gb[2] = { alloc((size_t)NG * D), alloc((size_t)NG * D) };
    float* mn2   = alloc(128);
    float* vr2   = alloc(128);

    // 536 MB A stream: TDM-staged tiles
    gemm_f(mapm, re_o, nullptr, pure_grid, NG, D, NR, 0, 1);

    for (int dir = 0; dir < 2; ++dir) {
        const int* esrc = (dir == 0 ? in_e : out_e);
        const int* edst = esrc + ET;
        fill0(deg, NG);
        k_deg<<<cdiv(ET, 256), 256, 0, stream>>>(edst, ew, deg, ET);
        k_dinv<<<cdiv(NG, 256), 256, 0, stream>>>(deg, dinv, NG);
        k_norm<<<cdiv(ET, 256), 256, 0, stream>>>(esrc, edst, ew, dinv, nrm, ET);
        const float* cur = pure_grid;
        for (int l = 0; l < 3; ++l) {
            float* dst = (l == 1) ? gb[dir] : ga[dir];        // 0->a, 1->b, 2->a
            gemm_f(cur, glinW  + (size_t)(dir * 3 + l) * D * D, nullptr, dst,   NG, D, D, 0, 0);
            gemm_f(cur, gconvW + (size_t)(dir * 3 + l) * D * D, nullptr, xlbuf, NG, D, D, 0, 0);
            fill0(hrbuf, NG * D);
            k_scat<<<cdiv(ET, 8), 256, 0, stream>>>(xlbuf, esrc, edst, nrm, hrbuf, ET, D);
            k_add<<<cdiv(NG * D, 256), 256, 0, stream>>>(dst, hrbuf, NG * D);
            k_bn_reduce<<<128, 256, 0, stream>>>(dst, mn2, vr2, NG, D);
            k_bn_apply<<<cdiv(NG * D, 256), 256, 0, stream>>>(dst, mn2, vr2,
                         gcbns + (dir * 3 + l) * D, gcbnb + (dir * 3 + l) * D,
                         NG, D, (l < 2) ? 1 : 0);
            cur = dst;
        }
    }
    float* full_cat = alloc((size_t)8193 * 256);
    k_full_cat<<<cdiv(8193 * 256, 256), 256, 0, stream>>>(ga[0], ga[1], full_cat, 8193 * 256);
    gemm_g(full_cat, fcW, fcb, fg_o, 8193, D, 256, 0);        // M=8193 -> generic

    // ---- RNNs ----
    float* temb = alloc((size_t)BL * 129);
    k_traces_emb<<<cdiv(BL * 129, 256), 256, 0, stream>>>(traces, fg_o, temb, BL * 129);
    float* bsum = alloc(768);
    k_bias_sums<<<3, 256, 0, stream>>>(tbih, tbhh, sbih, sbhh, bsum);
    float* XW = alloc((size_t)BL * D);
    float* y0 = alloc((size_t)BL * D);
    float* y1 = alloc((size_t)BL * D);
    float* y2 = alloc((size_t)BL * D);

    gemm_g(temb, tWih0, bsum + 0, XW, BL, D, 129, 0);         // K=129 -> generic
    rnn_recurrent<<<1, 256, 0, stream>>>(XW, tWhh, t_enc, y0, th_o);
    gemm_f(y0, tWih, bsum + 128, XW, BL, D, D, 0, 0);
    rnn_recurrent<<<1, 256, 0, stream>>>(XW, tWhh + 16384, t_enc + 4096, y1, th_o + 4096);
    gemm_f(y1, tWih + 16384, bsum + 256, XW, BL, D, D, 0, 0);
    rnn_recurrent<<<1, 256, 0, stream>>>(XW, tWhh + 32768, t_enc + 8192, y0, th_o + 8192);
    // y0 = traces_output

    float* semb = alloc((size_t)BL * D);
    k_gather_rows<<<cdiv(BL * D, 256), 256, 0, stream>>>(re_o, matched, semb, BL, D);
    gemm_f(semb, sWih, bsum + 384, XW, BL, D, D, 0, 0);
    rnn_recurrent<<<1, 256, 0, stream>>>(XW, sWhh, s_enc, y1, sh_o);
    gemm_f(y1, sWih + 16384, bsum + 512, XW, BL, D, D, 0, 0);
    rnn_recurrent<<<1, 256, 0, stream>>>(XW, sWhh + 16384, s_enc + 4096, y2, sh_o + 4096);
    gemm_f(y2, sWih + 32768, bsum + 640, XW, BL, D, D, 0, 0);
    rnn_recurrent<<<1, 256, 0, stream>>>(XW, sWhh + 32768, s_enc + 8192, y1, sh_o + 8192);
    // y1 = segments_output

    // ---- attention scoring ----
    float* tencb = alloc((size_t)BL * 64);
    float* sencb = alloc((size_t)BL * 64);
    gemm_f(y0, twW, twb, tencb, BL, 64, D, 0, 0);
    gemm_f(y1, swW, swb, sencb, BL, 64, D, 0, 0);
    float* comb = alloc((size_t)BL * D);
    k_comb<<<cdiv(BL * D, 256), 256, 0, stream>>>(tencb, sencb, comb, BL * D);
    float* xproj = alloc((size_t)BL * D);
    gemm_f(comb, attW, attb, xproj, BL, D, D, 2, 0);           // tanh
    float* cemb = alloc((size_t)BL * 10 * D);
    k_gather_rows<<<cdiv(BL * 10 * D, 256), 256, 0, stream>>>(re_o, cand, cemb, BL * 10, D);
    float* cproj = alloc((size_t)BL * 10 * D);
    gemm_f(cemb, atcW, atcb, cproj, BL * 10, D, D, 2, 0);      // tanh
    k_attn<<<cdiv(BL * 10, 256), 256, 0, stream>>>(xproj, cproj, av_o, BL * 10);
}